// GraphConvolutionTorch_45062796869923
// MI455X (gfx1250) — compile-verified
//
#include <hip/hip_runtime.h>
#include <hip/hip_bf16.h>

typedef float v2f __attribute__((ext_vector_type(2)));
typedef float v8f __attribute__((ext_vector_type(8)));

#define D 64  // D_IN == D_OUT == 64

// ---------------------------------------------------------------------------
// 1) zero the integer degree array
// ---------------------------------------------------------------------------
__global__ __launch_bounds__(256) void k_deg_init(int* __restrict__ degi, int N) {
    int i = blockIdx.x * 256 + threadIdx.x;
    if (i < N) degi[i] = 0;
}

// ---------------------------------------------------------------------------
// 2) in-degree on dst (self-loops folded in later as +1)
// ---------------------------------------------------------------------------
__global__ __launch_bounds__(256) void k_deg_count(const int* __restrict__ dst,
                                                   int* __restrict__ degi, int E) {
    int e = blockIdx.x * 256 + threadIdx.x;
    if (e < E) atomicAdd(&degi[dst[e]], 1);
}

// ---------------------------------------------------------------------------
// 3) dinv = rsqrt(deg + 1)   (+1 = self loop; always > 0)
// ---------------------------------------------------------------------------
__global__ __launch_bounds__(256) void k_dinv(const int* __restrict__ degi,
                                              float* __restrict__ dinv, int N) {
    int i = blockIdx.x * 256 + threadIdx.x;
    if (i < N) dinv[i] = rsqrtf((float)(degi[i] + 1));
}

// ---------------------------------------------------------------------------
// 4) y = dinv[i] * (x @ W^T)[i]  via V_WMMA_F32_16X16X4_F32
//    also seeds acc (== d_out) with y (the self-loop contribution).
//    One wave per 16x16 output tile; 8 waves (256 threads) per block.
//
//    A fragment (16x4 f32): lane L<16 holds A[L][k0+0..1] in v[0..1],
//                           lane L>=16 holds A[L-16][k0+2..3].
//    B fragment (4x16 f32): mirrored layout, N across lanes, K split by half.
//    B[k][n] = W[n][k]  (row-major W[out][in]) -> consecutive-k float2 loads.
//    C/D (16x16 f32): VGPR r, lane L -> row r + 8*(L>>4), col (L&15).
// ---------------------------------------------------------------------------
__global__ __launch_bounds__(256) void k_gemm_wmma(const float* __restrict__ x,
                                                   const float* __restrict__ W,
                                                   const float* __restrict__ dinv,
                                                   float* __restrict__ y,
                                                   float* __restrict__ acc,
                                                   int rowTiles) {
    const int lane = threadIdx.x & 31;
    const int wave = threadIdx.x >> 5;
    const int tile = blockIdx.x * 8 + wave;          // rowTiles*4 tiles total
    if (tile >= rowTiles * 4) return;                // wave-uniform guard

    const int row0 = (tile >> 2) * 16;
    const int col0 = (tile & 3) * 16;
    const int half = lane >> 4;                      // 0 or 1: K-half selector
    const int l15  = lane & 15;

    const float* arow = x + (size_t)(row0 + l15) * D;   // A row for this lane
    const float* brow = W + (size_t)(col0 + l15) * D;   // W row = B column

    v8f c = {};
#pragma unroll
    for (int k0 = 0; k0 < D; k0 += 4) {
        v2f a = *(const v2f*)(arow + k0 + 2 * half);
        v2f b = *(const v2f*)(brow + k0 + 2 * half);
        c = __builtin_amdgcn_wmma_f32_16x16x4_f32(
                /*neg_a=*/false, a, /*neg_b=*/false, b,
                /*c_mod=*/(short)0, c, /*reuse_a=*/false, /*reuse_b=*/false);
    }

#pragma unroll
    for (int r = 0; r < 8; ++r) {
        const int row = row0 + r + 8 * half;
        const float v = c[r] * dinv[row];
        const size_t o = (size_t)row * D + col0 + l15;
        y[o]   = v;   // gather source for scatter
        acc[o] = v;   // self-loop seed of the accumulator (acc == d_out)
    }
}

// ---------------------------------------------------------------------------
// 5) edge scatter: acc[dst] += y[src]; 16 lanes per edge, float4 per lane,
//    4 x global_atomic_add_f32 each. Coalesced 256B gather per edge.
// ---------------------------------------------------------------------------
__global__ __launch_bounds__(256) void k_scatter(const int* __restrict__ src,
                                                 const int* __restrict__ dst,
                                                 const float* __restrict__ y,
                                                 float* __restrict__ acc, int E) {
    const int gid = blockIdx.x * 256 + threadIdx.x;
    const int e = gid >> 4;
    if (e >= E) return;
    const int c = (gid & 15) << 2;                   // 0,4,...,60
    const int s = src[e];
    const int d = dst[e];
    const float4 v = *(const float4*)(y + (size_t)s * D + c);
    float* p = acc + (size_t)d * D + c;
    atomicAdd(p + 0, v.x);
    atomicAdd(p + 1, v.y);
    atomicAdd(p + 2, v.z);
    atomicAdd(p + 3, v.w);
}

// ---------------------------------------------------------------------------
// 6) out = relu(dinv[i] * acc + b[c])   (in place on d_out)
// ---------------------------------------------------------------------------
__global__ __launch_bounds__(256) void k_finalize(float* __restrict__ out,
                                                  const float* __restrict__ dinv,
                                                  const float* __restrict__ bias,
                                                  int total) {
    const int gid = blockIdx.x * 256 + threadIdx.x;
    if (gid >= total) return;
    const int row = gid >> 6;        // / D
    const int col = gid & (D - 1);
    const float v = fmaf(out[gid], dinv[row], bias[col]);
    out[gid] = fmaxf(v, 0.0f);
}

// ---------------------------------------------------------------------------
// launch
// ---------------------------------------------------------------------------
extern "C" void kernel_launch(void* const* d_in, const int* in_sizes, int n_in,
                              void* d_out, int out_size, void* d_ws, size_t ws_size,
                              hipStream_t stream) {
    const float* x    = (const float*)d_in[0];   // [N, 64]
    const float* W    = (const float*)d_in[1];   // [64, 64]
    const float* bias = (const float*)d_in[2];   // [64]
    const int*   edge = (const int*)d_in[3];     // [2, E] flat

    const int N = in_sizes[0] / D;               // 100000
    const int E = in_sizes[3] / 2;               // 1600000
    const int* src = edge;
    const int* dst = edge + E;

    float* out = (float*)d_out;                  // doubles as the accumulator

    // workspace layout: [degi: N i32][dinv: N f32][y: N*64 f32]
    char*  ws   = (char*)d_ws;
    int*   degi = (int*)ws;
    float* dinv = (float*)(ws + (size_t)N * 4);
    float* y    = (float*)(ws + (size_t)N * 8);

    const int rowTiles = N / 16;                 // N % 16 == 0 for this problem
    const int totTiles = rowTiles * 4;

    k_deg_init <<<(N + 255) / 256, 256, 0, stream>>>(degi, N);
    k_deg_count<<<(E + 255) / 256, 256, 0, stream>>>(dst, degi, E);
    k_dinv     <<<(N + 255) / 256, 256, 0, stream>>>(degi, dinv, N);
    k_gemm_wmma<<<(totTiles + 7) / 8, 256, 0, stream>>>(x, W, dinv, y, out, rowTiles);
    k_scatter  <<<((size_t)E * 16 + 255) / 256, 256, 0, stream>>>(src, dst, y, out, E);
    k_finalize <<<((size_t)N * D + 255) / 256, 256, 0, stream>>>(out, dinv, bias, N * D);
}